// SIN_LDS_VAE_37271726195457
// MI455X (gfx1250) — compile-verified
//
#include <hip/hip_runtime.h>
#include <math.h>

// ---------------------------------------------------------------------------
// Problem constants
// ---------------------------------------------------------------------------
#define Bn 16
#define Tn 512
#define Dn 64
#define NT 256          // threads per block (8 wave32)
#define LD 65           // padded LDS leading dim (bank-conflict free)
#define PLD 64          // unpadded stride for async-loaded tiles (16B rows)
#define MAT (LD * 64)   // padded matrix footprint in floats

#define LOG2PI_F 1.8378770664093453f

// d_out layout (floats), reference return order: z, kl, EX, EXXT, EXXNT
#define OFF_Z      0
#define OFF_KL     (Bn * Tn * Dn)                     // 524288
#define OFF_EX     (OFF_KL + Bn)                      // 524304
#define OFF_EXXT   (OFF_EX + Bn * Tn * Dn)            // 1048592
#define OFF_EXXNT  (OFF_EXXT + Bn * Tn * Dn * Dn)     // 34603024

// workspace layout (floats)
#define W_A       0
#define W_ATLAM   4096
#define W_ATLAMA  8192
#define W_LAM     12288
#define W_Q       16384
#define W_TAU     20480
#define W_B       24576
#define W_ATLAMB  24640
#define W_LAMB    24704
#define W_TAUMU   24768
#define W_SCAL    24832   // [0]=bTlamb [1]=logdet_lam [2]=c0
#define W_LOGZ    24848   // 16
#define W_SC1     24960   // 64x64 scratch
#define W_SC2     29056   // 64x64 scratch

// ---------------------------------------------------------------------------
// WMMA f32 16x16x4 fragment types
// ---------------------------------------------------------------------------
typedef float v2f __attribute__((ext_vector_type(2)));
typedef float v8f __attribute__((ext_vector_type(8)));

#if defined(__has_builtin)
#if __has_builtin(__builtin_amdgcn_wmma_f32_16x16x4_f32)
#define HAVE_WMMA_F32 1
#endif
#endif

// ---------------------------------------------------------------------------
// Async global->LDS streaming (CDNA5 GLOBAL_LOAD_ASYNC_TO_LDS_B128, ASYNCcnt)
// ---------------------------------------------------------------------------
#define WAIT_ASYNC(n) asm volatile("s_wait_asynccnt " #n ::: "memory")

// Copy one 64x64 f32 tile (row stride 64, rows 16B-aligned) into LDS without
// touching VGPR data paths. 1024 x B128 chunks; 4 chunks per thread, i.e.
// 4 async-load instructions per wave. Caller pairs with WAIT_ASYNC + barrier.
__device__ __forceinline__ void async_load_tile(const float* __restrict__ src,
                                                float* __restrict__ dst_lds,
                                                int tid) {
  const unsigned lds_base = (unsigned)(uintptr_t)dst_lds;  // low 32b = LDS addr
  for (int c = tid; c < 1024; c += NT) {
    const unsigned loff = lds_base + (unsigned)c * 16u;
    const float* g = src + (size_t)c * 4;
    asm volatile("global_load_async_to_lds_b128 %0, %1, off"
                 :
                 : "v"(loff), "v"(g)
                 : "memory");
  }
}

// ---------------------------------------------------------------------------
// 64x64x64 GEMM:  D = opA(A) * opB(B) [+ C],  optional software negation of A.
// All 256 threads participate; 8 waves x 2 output tiles; K split in 16 chunks
// of 4 feeding V_WMMA_F32_16X16X4_F32. Operands may live in LDS or global.
// ---------------------------------------------------------------------------
template <bool TA, bool TB, bool NEGA>
__device__ __forceinline__ void gemm64(float* D, int ldd,
                                       const float* A, int lda,
                                       const float* B, int ldb,
                                       const float* C, int ldc,
                                       int tid) {
  __syncthreads();
#if HAVE_WMMA_F32
  const int wave = tid >> 5;
  const int lane = tid & 31;
  const int half = lane >> 4;   // 0: lanes 0-15, 1: lanes 16-31
  const int lr   = lane & 15;
  for (int tile = wave; tile < 16; tile += 8) {
    const int mi = tile >> 2;
    const int ni = tile & 3;
    v8f acc;
#pragma unroll
    for (int v = 0; v < 8; ++v) {
      const int m = mi * 16 + half * 8 + v;
      const int n = ni * 16 + lr;
      acc[v] = C ? C[m * ldc + n] : 0.0f;
    }
#pragma unroll
    for (int kc = 0; kc < 16; ++kc) {
      const int kb = kc * 4 + half * 2;          // K pair for this lane half
      const int m  = mi * 16 + lr;
      const int n  = ni * 16 + lr;
      v2f a, bf;
      a.x  = TA ? A[(kb + 0) * lda + m] : A[m * lda + kb + 0];
      a.y  = TA ? A[(kb + 1) * lda + m] : A[m * lda + kb + 1];
      if (NEGA) { a.x = -a.x; a.y = -a.y; }      // f32 WMMA: NEG[1:0] must be 0
      bf.x = TB ? B[n * ldb + kb + 0] : B[(kb + 0) * ldb + n];
      bf.y = TB ? B[n * ldb + kb + 1] : B[(kb + 1) * ldb + n];
      acc = __builtin_amdgcn_wmma_f32_16x16x4_f32(
          false, a, false, bf, (short)0, acc, false, false);
    }
#pragma unroll
    for (int v = 0; v < 8; ++v) {
      const int m = mi * 16 + half * 8 + v;
      const int n = ni * 16 + lr;
      D[m * ldd + n] = acc[v];
    }
  }
#else
  // Scalar fallback (kept only so the file still compiles if the builtin
  // is absent; the WMMA path is the intended one).
  for (int idx = tid; idx < 4096; idx += NT) {
    const int i = idx >> 6, j = idx & 63;
    float s = C ? C[i * ldc + j] : 0.0f;
    for (int k = 0; k < 64; ++k) {
      const float a = TA ? A[k * lda + i] : A[i * lda + k];
      const float b = TB ? B[j * ldb + k] : B[k * ldb + j];
      s += (NEGA ? -a : a) * b;
    }
    D[i * ldd + j] = s;
  }
#endif
  __syncthreads();
}

// ---------------------------------------------------------------------------
// Block-cooperative dense linear-algebra helpers (VALU; 256 threads).
// Cholesky leaves L in lower triangle + diagonal; upper triangle is garbage.
// ---------------------------------------------------------------------------
__device__ __forceinline__ void chol64(float* M, int ld, int tid) {
  __syncthreads();
  for (int k = 0; k < 64; ++k) {
    if (tid == 0) M[k * ld + k] = sqrtf(M[k * ld + k]);
    __syncthreads();
    const float dinv = 1.0f / M[k * ld + k];
    for (int i = k + 1 + tid; i < 64; i += NT) M[i * ld + k] *= dinv;
    __syncthreads();
    const int n = 63 - k;
    for (int idx = tid; idx < n * n; idx += NT) {
      const int i = k + 1 + idx / n;
      const int j = k + 1 + idx % n;
      M[i * ld + j] -= M[i * ld + k] * M[j * ld + k];
    }
    __syncthreads();
  }
}

// Solve L * X = B in-place (B is 64x64).
__device__ __forceinline__ void trsm_lower64(const float* L, int ldl,
                                             float* B, int ldb, int tid) {
  __syncthreads();
  for (int r = 0; r < 64; ++r) {
    const float dinv = 1.0f / L[r * ldl + r];
    for (int j = tid; j < 64; j += NT) B[r * ldb + j] *= dinv;
    __syncthreads();
    for (int idx = tid; idx < (63 - r) * 64; idx += NT) {
      const int i = r + 1 + idx / 64;
      const int j = idx % 64;
      B[i * ldb + j] -= L[i * ldl + r] * B[r * ldb + j];
    }
    __syncthreads();
  }
}

// Solve L^T * X = B in-place (B is 64x64).
__device__ __forceinline__ void trsm_upperT64(const float* L, int ldl,
                                              float* B, int ldb, int tid) {
  __syncthreads();
  for (int r = 63; r >= 0; --r) {
    const float dinv = 1.0f / L[r * ldl + r];
    for (int j = tid; j < 64; j += NT) B[r * ldb + j] *= dinv;
    __syncthreads();
    for (int idx = tid; idx < r * 64; idx += NT) {
      const int i = idx / 64;
      const int j = idx % 64;
      B[i * ldb + j] -= L[r * ldl + i] * B[r * ldb + j];
    }
    __syncthreads();
  }
}

// Solve L * x = b in-place (x length 64).
__device__ __forceinline__ void trsv_lower64(const float* L, int ld,
                                             float* x, int tid) {
  __syncthreads();
  for (int r = 0; r < 64; ++r) {
    if (tid == 0) x[r] /= L[r * ld + r];
    __syncthreads();
    for (int i = r + 1 + tid; i < 64; i += NT) x[i] -= L[i * ld + r] * x[r];
    __syncthreads();
  }
}

// Solve L^T * x = b in-place (x length 64).
__device__ __forceinline__ void trsv_upperT64(const float* L, int ld,
                                              float* x, int tid) {
  __syncthreads();
  for (int r = 63; r >= 0; --r) {
    if (tid == 0) x[r] /= L[r * ld + r];
    __syncthreads();
    for (int i = tid; i < r; i += NT) x[i] -= L[r * ld + i] * x[r];
    __syncthreads();
  }
}

__device__ __forceinline__ float block_sum(float v, int tid, float* red) {
  red[tid] = v;
  __syncthreads();
  for (int s = NT / 2; s > 0; s >>= 1) {
    if (tid < s) red[tid] += red[tid + s];
    __syncthreads();
  }
  const float r = red[0];
  __syncthreads();
  return r;
}

// ---------------------------------------------------------------------------
// Kernel 1: one-time constant precompute into workspace.
// ---------------------------------------------------------------------------
__global__ void lds_setup_kernel(const float* __restrict__ mu,
                                 const float* __restrict__ Taup,
                                 const float* __restrict__ Lamp,
                                 const float* __restrict__ X,
                                 float* __restrict__ ws) {
  const int tid = threadIdx.x;
  __shared__ float red[NT];
  float* A    = ws + W_A;
  float* lam  = ws + W_LAM;
  float* tau  = ws + W_TAU;
  float* ATl  = ws + W_ATLAM;
  float* ATlA = ws + W_ATLAMA;
  float* bv   = ws + W_B;

  for (int idx = tid; idx < 4096; idx += NT) {
    const int i = idx >> 6, j = idx & 63;
    A[idx] = X[i * 65 + j];
  }
  for (int i = tid; i < 64; i += NT) bv[i] = X[i * 65 + 64];
  __syncthreads();

  for (int idx = tid; idx < 4096; idx += NT) {
    const int i = idx >> 6, j = idx & 63;
    float st = 0.0f, sl = 0.0f;
    for (int k = 0; k < 64; ++k) {
      st += Taup[i * 64 + k] * Taup[j * 64 + k];
      sl += Lamp[i * 64 + k] * Lamp[j * 64 + k];
    }
    tau[idx] = st + ((i == j) ? 1e-8f : 0.0f);
    lam[idx] = sl + ((i == j) ? 1e-8f : 0.0f);
  }
  __syncthreads();

  for (int idx = tid; idx < 4096; idx += NT) {
    const int i = idx >> 6, j = idx & 63;
    float s = 0.0f;
    for (int k = 0; k < 64; ++k) s += A[k * 64 + i] * lam[k * 64 + j];
    ATl[idx] = s;
  }
  __syncthreads();

  for (int idx = tid; idx < 4096; idx += NT) {
    const int i = idx >> 6, j = idx & 63;
    float s = 0.0f;
    for (int k = 0; k < 64; ++k) s += ATl[i * 64 + k] * A[k * 64 + j];
    ATlA[idx] = s;
  }
  for (int i = tid; i < 64; i += NT) {
    float s1 = 0.0f, s2 = 0.0f, s3 = 0.0f;
    for (int k = 0; k < 64; ++k) {
      s1 += ATl[i * 64 + k] * bv[k];
      s2 += lam[i * 64 + k] * bv[k];
      s3 += tau[i * 64 + k] * mu[k];
    }
    ws[W_ATLAMB + i] = s1;
    ws[W_LAMB + i]   = s2;
    ws[W_TAUMU + i]  = s3;
  }
  __syncthreads();

  float p = 0.0f;
  for (int i = tid; i < 64; i += NT) p += bv[i] * ws[W_LAMB + i];
  const float bTlamb = block_sum(p, tid, red);

  float* SC1 = ws + W_SC1;
  float* SC2 = ws + W_SC2;
  for (int idx = tid; idx < 4096; idx += NT) SC1[idx] = lam[idx];
  chol64(SC1, 64, tid);
  p = 0.0f;
  for (int i = tid; i < 64; i += NT) p += 2.0f * logf(SC1[i * 64 + i]);
  const float logdet_lam = block_sum(p, tid, red);

  for (int idx = tid; idx < 4096; idx += NT) {
    const int i = idx >> 6, j = idx & 63;
    SC2[idx] = (i == j) ? 1.0f : 0.0f;
  }
  trsm_lower64(SC1, 64, SC2, 64, tid);                       // SC2 = L^{-1}
  gemm64<true, false, false>(ws + W_Q, 64, SC2, 64, SC2, 64, // Q = V^T V
                             (const float*)nullptr, 0, tid);

  for (int idx = tid; idx < 4096; idx += NT) SC1[idx] = tau[idx];
  chol64(SC1, 64, tid);
  p = 0.0f;
  for (int i = tid; i < 64; i += NT) p += 2.0f * logf(SC1[i * 64 + i]);
  const float logdet_tau = block_sum(p, tid, red);
  p = 0.0f;
  for (int i = tid; i < 64; i += NT) p += mu[i] * ws[W_TAUMU + i];
  const float mtm = block_sum(p, tid, red);

  if (tid == 0) {
    ws[W_SCAL + 0] = bTlamb;
    ws[W_SCAL + 1] = logdet_lam;
    ws[W_SCAL + 2] = -0.5f * mtm + 0.5f * logdet_tau - 0.5f * 64.0f * LOG2PI_F;
  }
}

// ---------------------------------------------------------------------------
// Kernel 2: forward information filter, fused with moment extraction.
// One block per batch chain. Emits m_f -> EX region, P_f -> EXXT region,
// logZ -> workspace. P_f = V^T V (WMMA syrk), J update = lam+Jr - W^T W (WMMA).
// ---------------------------------------------------------------------------
__global__ void lds_forward_kernel(const float* __restrict__ Jr,
                                   const float* __restrict__ hr,
                                   float* __restrict__ ws,
                                   float* __restrict__ out) {
  extern __shared__ float sm[];
  float* J   = sm;
  float* Jc  = sm + MAT;
  float* W   = sm + 2 * MAT;
  float* h   = sm + 3 * MAT;
  float* uv  = h + 64;
  float* jv  = uv + 64;
  float* mv  = jv + 64;
  float* t1  = mv + 64;
  float* red = t1 + 64;
  float* sc  = red + NT;   // sc[0] = running log-normalizer c

  const int b   = blockIdx.x;
  const int tid = threadIdx.x;
  const float* Jr_b = Jr + (size_t)b * Tn * 4096;
  const float* hr_b = hr + (size_t)b * Tn * 64;
  float* EX   = out + OFF_EX + (size_t)b * Tn * 64;
  float* EXXT = out + OFF_EXXT + (size_t)b * Tn * 4096;

  // J0 = tau + Jr[0],  h0 = tau_mu + hr[0],  c = c0
  for (int idx = tid; idx < 4096; idx += NT) {
    const int i = idx >> 6, j = idx & 63;
    J[i * LD + j] = ws[W_TAU + idx] + Jr_b[idx];
  }
  for (int i = tid; i < 64; i += NT) h[i] = ws[W_TAUMU + i] + hr_b[i];
  if (tid == 0) sc[0] = ws[W_SCAL + 2];
  __syncthreads();

  for (int t = 0; t < Tn; ++t) {
    if (t < Tn - 1) {  // prefetch next step's precision block
      for (int r = tid; r < 64; r += NT)
        __builtin_prefetch(Jr_b + (size_t)(t + 1) * 4096 + r * 64, 0, 1);
    }
    // ---- moments at t:  L = chol(J_f[t]); V = L^{-1}; P = V^T V; m = P h
    for (int idx = tid; idx < 4096; idx += NT) {
      const int i = idx >> 6, j = idx & 63;
      Jc[i * LD + j] = J[i * LD + j];
    }
    chol64(Jc, LD, tid);
    for (int idx = tid; idx < 4096; idx += NT) {
      const int i = idx >> 6, j = idx & 63;
      W[i * LD + j] = (i == j) ? 1.0f : 0.0f;
    }
    trsm_lower64(Jc, LD, W, LD, tid);                 // W = V (lower)
    for (int i = tid; i < 64; i += NT) {
      float s = 0.0f;
      for (int k = 0; k < 64; ++k) s += W[i * LD + k] * h[k];
      t1[i] = s;
    }
    __syncthreads();
    for (int i = tid; i < 64; i += NT) {
      float s = 0.0f;
      for (int k = 0; k < 64; ++k) s += W[k * LD + i] * t1[k];
      mv[i] = s;
      EX[(size_t)t * 64 + i] = s;
    }
    __syncthreads();
    gemm64<true, false, false>(EXXT + (size_t)t * 4096, 64, W, LD, W, LD,
                               (const float*)nullptr, 0, tid);   // P_f[t]

    if (t == Tn - 1) {
      float p = 0.0f;
      for (int i = tid; i < 64; i += NT) p += 2.0f * logf(Jc[i * LD + i]);
      const float ldJ = block_sum(p, tid, red);
      p = 0.0f;
      for (int i = tid; i < 64; i += NT) p += h[i] * mv[i];
      const float hm = block_sum(p, tid, red);
      if (tid == 0)
        ws[W_LOGZ + b] = sc[0] + 0.5f * (64.0f * LOG2PI_F - ldJ + hm);
      break;
    }

    // ---- filter step:  Jtt = J + A^T lam A ; Lt = chol(Jtt)
    for (int idx = tid; idx < 4096; idx += NT) {
      const int i = idx >> 6, j = idx & 63;
      Jc[i * LD + j] = J[i * LD + j] + ws[W_ATLAMA + idx];
    }
    chol64(Jc, LD, tid);
    float pd = 0.0f;
    for (int i = tid; i < 64; i += NT) pd += 2.0f * logf(Jc[i * LD + i]);
    const float ld_t = block_sum(pd, tid, red);

    // u = h - A^T lam b ;  y = Lt^{-1} u ;  Jinv_u = Lt^{-T} y
    for (int i = tid; i < 64; i += NT) uv[i] = h[i] - ws[W_ATLAMB + i];
    trsv_lower64(Jc, LD, uv, tid);
    float py = 0.0f;
    for (int i = tid; i < 64; i += NT) py += uv[i] * uv[i];
    const float ydy = block_sum(py, tid, red);
    for (int i = tid; i < 64; i += NT) jv[i] = uv[i];
    trsv_upperT64(Jc, LD, jv, tid);
    if (tid == 0)
      sc[0] += 0.5f * (ws[W_SCAL + 1] - ld_t - ws[W_SCAL + 0] + ydy);

    // W = Lt^{-1} (A^T lam)  ;  J_{t+1} = lam + Jr[t+1] - W^T W
    for (int idx = tid; idx < 4096; idx += NT) {
      const int i = idx >> 6, j = idx & 63;
      W[i * LD + j] = ws[W_ATLAM + idx];
    }
    trsm_lower64(Jc, LD, W, LD, tid);
    for (int idx = tid; idx < 4096; idx += NT) {
      const int i = idx >> 6, j = idx & 63;
      Jc[i * LD + j] = ws[W_LAM + idx] + Jr_b[(size_t)(t + 1) * 4096 + idx];
    }
    gemm64<true, false, true>(J, LD, W, LD, W, LD, Jc, LD, tid);

    // h_{t+1} = lam b + lam A Jinv_u + hr[t+1]   (lam A = (A^T lam)^T)
    for (int i = tid; i < 64; i += NT) {
      float s = ws[W_LAMB + i] + hr_b[(size_t)(t + 1) * 64 + i];
      for (int k = 0; k < 64; ++k) s += ws[W_ATLAM + k * 64 + i] * jv[k];
      h[i] = s;
    }
    __syncthreads();
  }
}

// ---------------------------------------------------------------------------
// Kernel 3: RTS smoother + backward sampler + KL. One block per chain.
// P_f[t] tiles stream into LDS via double-buffered GLOBAL_LOAD_ASYNC_TO_LDS
// (ASYNCcnt), overlapping the next tile's fetch with this step's 5 WMMA GEMMs.
// G is kept as X = G^T (X = P_pred^{-1} A P_t) and applied via transpose flags.
// ---------------------------------------------------------------------------
__global__ void lds_backward_kernel(const float* __restrict__ Jr,
                                    const float* __restrict__ hr,
                                    const float* __restrict__ eps,
                                    const float* __restrict__ ws,
                                    float* __restrict__ out) {
  extern __shared__ float sm[];
  float* A_s = sm;
  float* Pn  = sm + 1 * MAT;
  float* T1  = sm + 2 * MAT;
  float* Xg  = sm + 3 * MAT;  // X = G^T
  float* GPn = sm + 4 * MAT;
  float* SCR = sm + 5 * MAT;  // Ppred/Lp -> GT1 -> C/Lc
  float* PtA = sm + 6 * MAT;          // async-filled, stride PLD=64
  float* PtB = sm + 6 * MAT + 4096;   // async-filled, stride PLD=64
  float* vec = sm + 6 * MAT + 8192;
  float* m_t   = vec;
  float* m_n   = vec + 64;
  float* m_s   = vec + 128;
  float* x_n   = vec + 192;
  float* dv    = vec + 256;
  float* tv    = vec + 320;
  float* epss  = vec + 384;
  float* mpred = vec + 448;
  float* red   = vec + 512;

  const int b   = blockIdx.x;
  const int tid = threadIdx.x;
  const float* Jr_b  = Jr + (size_t)b * Tn * 4096;
  const float* hr_b  = hr + (size_t)b * Tn * 64;
  const float* eps_b = eps + (size_t)b * Tn * 64;
  float* z     = out + OFF_Z + (size_t)b * Tn * 64;
  float* EX    = out + OFF_EX + (size_t)b * Tn * 64;
  float* EXXT  = out + OFF_EXXT + (size_t)b * Tn * 4096;
  float* EXXNT = out + OFF_EXXNT + (size_t)b * (Tn - 1) * 4096;
  const float* Qg = ws + W_Q;

  float pJ = 0.0f, pH = 0.0f;  // kl accumulators

  // Kick off the async stream for the first smoother tile P_f[T-2].
  async_load_tile(EXXT + (size_t)(Tn - 2) * 4096, PtA, tid);

  // constants + terminal state
  for (int idx = tid; idx < 4096; idx += NT) {
    const int i = idx >> 6, j = idx & 63;
    A_s[i * LD + j] = ws[W_A + idx];
    Pn[i * LD + j]  = EXXT[(size_t)(Tn - 1) * 4096 + idx];   // P_f[T-1]
  }
  for (int i = tid; i < 64; i += NT) {
    m_n[i]  = EX[(size_t)(Tn - 1) * 64 + i];                  // m_f[T-1]
    epss[i] = eps_b[(size_t)(Tn - 1) * 64 + i];
  }
  __syncthreads();
  for (int idx = tid; idx < 4096; idx += NT) {
    const int i = idx >> 6, j = idx & 63;
    SCR[i * LD + j] = Pn[i * LD + j] + ((i == j) ? 1e-6f : 0.0f);
  }
  chol64(SCR, LD, tid);
  for (int i = tid; i < 64; i += NT) {
    float s = m_n[i];
    for (int k = 0; k <= i; ++k) s += SCR[i * LD + k] * epss[k];
    x_n[i] = s;
    z[(size_t)(Tn - 1) * 64 + i] = s;
    pH += hr_b[(size_t)(Tn - 1) * 64 + i] * m_n[i];
  }
  __syncthreads();
  for (int idx = tid; idx < 4096; idx += NT) {
    const int i = idx >> 6, j = idx & 63;
    const float v = Pn[i * LD + j] + m_n[i] * m_n[j];
    EXXT[(size_t)(Tn - 1) * 4096 + idx] = v;
    pJ += Jr_b[(size_t)(Tn - 1) * 4096 + idx] * v;
  }
  __syncthreads();

  for (int t = Tn - 2; t >= 0; --t) {
    float* Ptc = ((Tn - 2 - t) & 1) ? PtB : PtA;  // tile for this step
    float* Ptn = ((Tn - 2 - t) & 1) ? PtA : PtB;  // tile being prefetched

    // 1. issue next tile, then drain this step's 4 async ops per wave.
    if (t > 0) {
      async_load_tile(EXXT + (size_t)(t - 1) * 4096, Ptn, tid);
      WAIT_ASYNC(0x4);
      for (int r = tid; r < 64; r += NT)
        __builtin_prefetch(Jr_b + (size_t)(t - 1) * 4096 + r * 64, 0, 1);
    } else {
      WAIT_ASYNC(0x0);
    }
    for (int i = tid; i < 64; i += NT) {
      m_t[i]  = EX[(size_t)t * 64 + i];
      epss[i] = eps_b[(size_t)t * 64 + i];
    }
    __syncthreads();

    // 2-3. T1 = A P_t ; P_pred = T1 A^T + Q
    gemm64<false, false, false>(T1, LD, A_s, LD, Ptc, PLD,
                                (const float*)nullptr, 0, tid);
    gemm64<false, true, false>(SCR, LD, T1, LD, A_s, LD, Qg, 64, tid);

    // 4. m_pred = A m_t + b
    for (int i = tid; i < 64; i += NT) {
      float s = ws[W_B + i];
      for (int k = 0; k < 64; ++k) s += A_s[i * LD + k] * m_t[k];
      mpred[i] = s;
    }
    __syncthreads();

    // 5-6. Lp = chol(P_pred) ; X = Lp^{-T} Lp^{-1} T1   (G = X^T)
    chol64(SCR, LD, tid);
    for (int idx = tid; idx < 4096; idx += NT) {
      const int i = idx >> 6, j = idx & 63;
      Xg[i * LD + j] = T1[i * LD + j];
    }
    trsm_lower64(SCR, LD, Xg, LD, tid);
    trsm_upperT64(SCR, LD, Xg, LD, tid);

    // 7. m_s = m_t + G (m_n - m_pred)
    for (int i = tid; i < 64; i += NT) dv[i] = m_n[i] - mpred[i];
    __syncthreads();
    for (int i = tid; i < 64; i += NT) {
      float s = 0.0f;
      for (int k = 0; k < 64; ++k) s += Xg[k * LD + i] * dv[k];
      m_s[i] = m_t[i] + s;
    }
    __syncthreads();

    // 8. GPn = G P_next = X^T P_n
    gemm64<true, false, false>(GPn, LD, Xg, LD, Pn, LD,
                               (const float*)nullptr, 0, tid);

    // 9. EXXNT[t] = GPn + m_s m_n^T
    for (int idx = tid; idx < 4096; idx += NT) {
      const int i = idx >> 6, j = idx & 63;
      EXXNT[(size_t)t * 4096 + idx] = GPn[i * LD + j] + m_s[i] * m_n[j];
    }
    __syncthreads();

    // 10. M1 = GPn - T1^T   (G(P_next - P_pred), using G P_pred = T1^T)
    for (int idx = tid; idx < 4096; idx += NT) {
      const int i = idx >> 6, j = idx & 63;
      GPn[i * LD + j] -= T1[j * LD + i];
    }
    // 11. P_s = M1 G^T + P_t  -> becomes new carry in Pn
    gemm64<false, false, false>(Pn, LD, GPn, LD, Xg, LD, Ptc, PLD, tid);
    // 12. GT1 = G T1 = X^T T1   (for C = P_t - (G T1)^T)
    gemm64<true, false, false>(SCR, LD, Xg, LD, T1, LD,
                               (const float*)nullptr, 0, tid);

    // 13. outputs + kl terms at t
    for (int idx = tid; idx < 4096; idx += NT) {
      const int i = idx >> 6, j = idx & 63;
      const float v = Pn[i * LD + j] + m_s[i] * m_s[j];
      EXXT[(size_t)t * 4096 + idx] = v;
      pJ += Jr_b[(size_t)t * 4096 + idx] * v;
    }
    for (int i = tid; i < 64; i += NT) {
      EX[(size_t)t * 64 + i] = m_s[i];
      pH += hr_b[(size_t)t * 64 + i] * m_s[i];
    }
    __syncthreads();

    // 14. C = P_t - GT1^T + 1e-6 I  (pairwise, in-place in SCR, race-free)
    for (int idx = tid; idx < 2080; idx += NT) {
      int i = (int)((sqrtf(8.0f * (float)idx + 1.0f) - 1.0f) * 0.5f);
      while ((i + 1) * (i + 2) / 2 <= idx) ++i;
      while (i * (i + 1) / 2 > idx) --i;
      const int j = idx - i * (i + 1) / 2;  // j <= i
      const float lo = Ptc[i * PLD + j] - SCR[j * LD + i];
      const float hi = Ptc[j * PLD + i] - SCR[i * LD + j];
      SCR[i * LD + j] = lo + ((i == j) ? 1e-6f : 0.0f);
      if (i != j) SCR[j * LD + i] = hi;
    }
    // 15. Lc = chol(C)
    chol64(SCR, LD, tid);

    // 16. x_t = m_t + G (x_n - m_pred) + Lc eps_t
    for (int i = tid; i < 64; i += NT) dv[i] = x_n[i] - mpred[i];
    __syncthreads();
    for (int i = tid; i < 64; i += NT) {
      float s = 0.0f;
      for (int k = 0; k < 64; ++k) s += Xg[k * LD + i] * dv[k];
      tv[i] = s;
    }
    __syncthreads();
    for (int i = tid; i < 64; i += NT) {
      float s = m_t[i] + tv[i];
      for (int k = 0; k <= i; ++k) s += SCR[i * LD + k] * epss[k];
      x_n[i] = s;
      z[(size_t)t * 64 + i] = s;
    }
    // 17. carry m_n = m_s
    __syncthreads();
    for (int i = tid; i < 64; i += NT) m_n[i] = m_s[i];
    __syncthreads();
  }

  // kl[b] = -0.5 * <J_r, EXXT> + <h_r, EX> - logZ
  const float tot = block_sum(-0.5f * pJ + pH, tid, red);
  if (tid == 0) out[OFF_KL + b] = tot - ws[W_LOGZ + b];
}

// ---------------------------------------------------------------------------
// Launch
// ---------------------------------------------------------------------------
extern "C" void kernel_launch(void* const* d_in, const int* in_sizes, int n_in,
                              void* d_out, int out_size, void* d_ws,
                              size_t ws_size, hipStream_t stream) {
  (void)in_sizes; (void)n_in; (void)out_size; (void)ws_size;
  const float* Jr   = (const float*)d_in[0];
  const float* hr   = (const float*)d_in[1];
  const float* eps  = (const float*)d_in[2];
  const float* mu   = (const float*)d_in[3];
  const float* Taup = (const float*)d_in[4];
  const float* Lamp = (const float*)d_in[5];
  const float* X    = (const float*)d_in[6];
  float* out = (float*)d_out;
  float* ws  = (float*)d_ws;

  const size_t fwd_smem = (size_t)(3 * MAT + 5 * 64 + NT + 8) * sizeof(float);
  const size_t bwd_smem = (size_t)(6 * MAT + 2 * 4096 + 8 * 64 + NT) * sizeof(float);

  lds_setup_kernel<<<1, NT, 0, stream>>>(mu, Taup, Lamp, X, ws);
  lds_forward_kernel<<<Bn, NT, fwd_smem, stream>>>(Jr, hr, ws, out);
  lds_backward_kernel<<<Bn, NT, bwd_smem, stream>>>(Jr, hr, eps, ws, out);
}